// LightGCN_14748917694876
// MI455X (gfx1250) — compile-verified
//
#include <hip/hip_runtime.h>

typedef __attribute__((ext_vector_type(2))) float v2f;
typedef __attribute__((ext_vector_type(8))) float v8f;

#define DIM 64
#define LAYERS 3

// ---------------------------------------------------------------------------
// init: h = xk = concat(user_emb, item_emb)   (float4-vectorized; DIM*rows % 4 == 0)
// ---------------------------------------------------------------------------
__global__ void lgcn_init(const float* __restrict__ user_emb,
                          const float* __restrict__ item_emb,
                          float* __restrict__ h, float* __restrict__ xk,
                          long n_user_el, long n_total_vec4) {
  long i = (long)blockIdx.x * blockDim.x + threadIdx.x;
  if (i >= n_total_vec4) return;
  long e = i * 4;
  const float4 v = (e < n_user_el)
      ? *reinterpret_cast<const float4*>(user_emb + e)
      : *reinterpret_cast<const float4*>(item_emb + (e - n_user_el));
  *reinterpret_cast<float4*>(h + e)  = v;
  *reinterpret_cast<float4*>(xk + e) = v;
}

// ---------------------------------------------------------------------------
// COO SpMM scatter: y[row] += val * x[col]  (vector of 64 floats per edge)
// 16 lanes per edge, float4 per lane: one contiguous 256B gather segment and
// one contiguous 256B atomic segment per edge; x/y live in L2 (38.4MB each).
// ---------------------------------------------------------------------------
__global__ void lgcn_scatter(const int* __restrict__ rows,
                             const int* __restrict__ cols,
                             const float* __restrict__ vals,
                             const float* __restrict__ x,
                             float* __restrict__ y, int nnz) {
  long tid  = (long)blockIdx.x * blockDim.x + threadIdx.x;
  int edge  = (int)(tid >> 4);
  int part  = (int)(tid & 15);
  if (edge >= nnz) return;
  int   r = rows[edge];
  int   c = cols[edge];
  float v = vals[edge];
  const float4 xv = *reinterpret_cast<const float4*>(x + (size_t)c * DIM + part * 4);
  float* dst = y + (size_t)r * DIM + part * 4;
  atomicAdd(dst + 0, v * xv.x);
  atomicAdd(dst + 1, v * xv.y);
  atomicAdd(dst + 2, v * xv.z);
  atomicAdd(dst + 3, v * xv.w);
}

// ---------------------------------------------------------------------------
// h += y   (float4-vectorized)
// ---------------------------------------------------------------------------
__global__ void lgcn_accum(float* __restrict__ h, const float* __restrict__ y,
                           long n_vec4) {
  long i = (long)blockIdx.x * blockDim.x + threadIdx.x;
  if (i >= n_vec4) return;
  long e = i * 4;
  float4 a = *reinterpret_cast<float4*>(h + e);
  float4 b = *reinterpret_cast<const float4*>(y + e);
  a.x += b.x; a.y += b.y; a.z += b.z; a.w += b.w;
  *reinterpret_cast<float4*>(h + e) = a;
}

// ---------------------------------------------------------------------------
// Batched dot products via V_WMMA_F32_16X16X4_F32.
// One wave handles 16 (user,item) pairs: A = H_u rows (16x64), B = H_i^T
// (64x16); 16 chained K=4 WMMAs accumulate D = A*B in f32; diagonal D[i][i]
// is the i-th dot product. Output scaled by 1/16 = (1/(LAYERS+1))^2.
//
// Operand layouts (ISA 7.12.2, 32-bit, wave32):
//   A 16x4 : lane m=lane&15 holds row m; lane-half selects K base 0 or 2;
//            VGPR0=K+0, VGPR1=K+1  -> per-lane float2 from hu[m][k0+kb ..]
//   B 4x16 : lane n=lane&15 holds col n; same K mapping -> float2 from hi[n]
//   C/D    : D[i][i] in lane i (i<8, c[i]) and lane i+16 (i>=8, c[i-8])
// ---------------------------------------------------------------------------
__global__ void lgcn_dot_wmma(const int* __restrict__ users,
                              const int* __restrict__ items,
                              const float* __restrict__ h,
                              float* __restrict__ out,
                              int n_users, int n_groups) {
  const int lane  = threadIdx.x & 31;
  const int wave  = threadIdx.x >> 5;
  int group = blockIdx.x * (blockDim.x >> 5) + wave;
  if (group >= n_groups) group = 0;   // duplicate work, identical value: no divergence, EXEC stays all-1s

  const int m  = lane & 15;           // A row / B column handled by this lane
  const int kb = (lane >> 4) << 1;    // K base within each K=4 step: 0 or 2
  const int b  = group * 16 + m;

  const int u  = users[b];
  const int it = items[b];
  const float* __restrict__ hu = h + (size_t)u * DIM;
  const float* __restrict__ hi = h + (size_t)(n_users + it) * DIM;

  v8f c = {};
#pragma unroll
  for (int k0 = 0; k0 < DIM; k0 += 4) {
    v2f a, bb;
    a.x  = hu[k0 + kb];
    a.y  = hu[k0 + kb + 1];
    bb.x = hi[k0 + kb];
    bb.y = hi[k0 + kb + 1];
    // (neg_a, A, neg_b, B, c_mod, C, reuse_a, reuse_b)
    c = __builtin_amdgcn_wmma_f32_16x16x4_f32(false, a, false, bb,
                                              (short)0, c, false, false);
  }

  // Extract the diagonal.
  float r = 0.0f;
#pragma unroll
  for (int q = 0; q < 8; ++q) {
    if (lane == q)      r = c[q];   // D[q][q],     q = 0..7
    if (lane == 24 + q) r = c[q];   // D[8+q][8+q], q = 0..7
  }
  const float scale = 1.0f / 16.0f; // (1/(LAYERS+1))^2 folded into the product
  if (lane < 8)        out[group * 16 + lane]        = r * scale;
  else if (lane >= 24) out[group * 16 + (lane - 16)] = r * scale;
}

// ---------------------------------------------------------------------------
// Host-side orchestration (graph-capturable: kernels + hipMemsetAsync only)
// ---------------------------------------------------------------------------
extern "C" void kernel_launch(void* const* d_in, const int* in_sizes, int n_in,
                              void* d_out, int out_size, void* d_ws, size_t ws_size,
                              hipStream_t stream) {
  const int*   users    = (const int*)d_in[0];
  const int*   items    = (const int*)d_in[1];
  const int*   A_rows   = (const int*)d_in[2];
  const int*   A_cols   = (const int*)d_in[3];
  const float* A_vals   = (const float*)d_in[4];
  const float* user_emb = (const float*)d_in[5];
  const float* item_emb = (const float*)d_in[6];
  float*       out      = (float*)d_out;

  const int  batch      = in_sizes[0];
  const int  nnz        = in_sizes[4];
  const long n_user_el  = (long)in_sizes[5];
  const long n_item_el  = (long)in_sizes[6];
  const long n_total_el = n_user_el + n_item_el;      // 150000 * 64
  const int  n_users    = (int)(n_user_el / DIM);

  // Workspace: h | x0 | x1, each n_total_el floats (3 * 38.4MB = 115.2MB,
  // all resident in the 192MB L2 together with the edge stream).
  float* h  = (float*)d_ws;
  float* x0 = h  + n_total_el;
  float* x1 = x0 + n_total_el;

  const int  T = 256;
  const long nvec4 = n_total_el / 4;

  lgcn_init<<<(int)((nvec4 + T - 1) / T), T, 0, stream>>>(
      user_emb, item_emb, h, x0, n_user_el, nvec4);

  float* cur = x0;
  float* nxt = x1;
  for (int layer = 0; layer < LAYERS; ++layer) {
    hipMemsetAsync(nxt, 0, (size_t)n_total_el * sizeof(float), stream);
    long sthreads = (long)nnz * 16;
    lgcn_scatter<<<(int)((sthreads + T - 1) / T), T, 0, stream>>>(
        A_rows, A_cols, A_vals, cur, nxt, nnz);
    lgcn_accum<<<(int)((nvec4 + T - 1) / T), T, 0, stream>>>(h, nxt, nvec4);
    float* t = cur; cur = nxt; nxt = t;
  }

  const int n_groups = batch / 16;            // 1024
  const int waves_pb = T / 32;                // 8 waves per block
  const int blocks   = (n_groups + waves_pb - 1) / waves_pb;
  lgcn_dot_wmma<<<blocks, T, 0, stream>>>(users, items, h, out, n_users, n_groups);
}